// PDP_70351564308621
// MI455X (gfx1250) — compile-verified
//
#include <hip/hip_runtime.h>
#include <stdint.h>

// ---------------------------------------------------------------------------
// PDP unstructured pruning mask: exact radix-select of two adjacent order
// statistics of |w| over 67M elements, then elementwise sigmoid gating.
//
// Pass 1 uses the CDNA5 Tensor Data Mover: wave 0 of each workgroup DMAs
// 32 KB tiles of the weight into LDS (tensor_load_to_lds, TENSORcnt),
// double-buffered against the LDS-atomic histogram done by all 16 waves.
//
// ws layout (uint32 words):
//   [0 .. 32767]                    hist_hi  (top-15 bits of |w| bit pattern)
//   [32768 .. 32768+2*65536-1]      hist_lo[2] (low-16 bits, per target bucket)
//   [WS_CTL ..]                     control: 0:h16_A 1:remA 2:h16_B 3:remB
//                                            4:keyA 5:keyB 6:t^2 (float bits)
// ---------------------------------------------------------------------------

typedef float f32x4 __attribute__((ext_vector_type(4)));
typedef unsigned int u32x4 __attribute__((ext_vector_type(4)));
typedef int i32x8 __attribute__((ext_vector_type(8)));
typedef int i32x4 __attribute__((ext_vector_type(4)));

#define NB_HI 32768
#define NB_LO 65536
#define WS_H1 0
#define WS_H2 (NB_HI)
#define WS_CTL (NB_HI + 2 * NB_LO)
#define WS_WORDS (WS_CTL + 16)

#define TILE_ELEMS 8192                 // 32 KB per tile
#define HIST_SMEM_BYTES (NB_HI * 4 + 2 * TILE_ELEMS * 4)  // 192 KB (<320 KB/WGP)

#if defined(__HIP_DEVICE_COMPILE__) && __has_builtin(__builtin_amdgcn_tensor_load_to_lds) && \
    __has_builtin(__builtin_amdgcn_s_wait_tensorcnt)
#define USE_TDM 1
#else
#define USE_TDM 0
#endif

__global__ __launch_bounds__(256) void k_zero(uint32_t* __restrict__ ws, int nwords) {
  int i = blockIdx.x * blockDim.x + threadIdx.x;
  int s = gridDim.x * blockDim.x;
  for (; i < nwords; i += s) ws[i] = 0u;
}

#if USE_TDM
// Issue a 1-D TDM tile load: tile_elems fp32 words, global -> LDS byte offset.
__device__ __forceinline__ void tdm_load_1d(const float* gaddr, uint32_t lds_off,
                                            uint32_t tile_elems) {
  uint64_t ga = (uint64_t)(uintptr_t)gaddr;
  u32x4 g0;
  g0.x = 1u;                                   // count=1 (valid user descriptor)
  g0.y = lds_off;                              // lds_addr (bytes)
  g0.z = (uint32_t)ga;                         // global_addr[31:0]
  g0.w = (uint32_t)(ga >> 32) | (2u << 30);    // global_addr[56:32] | type=2
  i32x8 g1;
  g1[0] = (int)(2u << 16);                     // workgroup_mask=0, data_size=2 (4B)
  g1[1] = (int)((tile_elems & 0xFFFFu) << 16); // tensor_dim0[15:0] (== tile span)
  g1[2] = (int)((tile_elems >> 16) & 0xFFFFu); // tensor_dim0[31:16], tensor_dim1=0
  g1[3] = (int)((tile_elems & 0xFFFFu) << 16); // tile_dim0
  g1[4] = 0;                                   // tile_dim1=0, tile_dim2=0 (1-D)
  g1[5] = 0;                                   // tensor_dim0_stride lo (unused, 1-D)
  g1[6] = 0;
  g1[7] = 0;
  i32x4 z4 = {0, 0, 0, 0};                     // groups 2/3 unused (<=2D tensor)
  i32x8 z8 = {0, 0, 0, 0, 0, 0, 0, 0};         // extra group (6-arg toolchain), unused
  __builtin_amdgcn_tensor_load_to_lds(g0, g1, z4, z4, z8, 0);
}
#endif

// Pass 1: LDS-privatized histogram of key>>16 (key = bits(|w|), 15 useful bits)
__global__ __launch_bounds__(512)
void k_hist_hi(const float* __restrict__ w, int n, uint32_t* __restrict__ gh) {
  extern __shared__ uint32_t smem[];
  uint32_t* lh = smem;                          // NB_HI counters (128 KB)
  for (int b = threadIdx.x; b < NB_HI; b += blockDim.x) lh[b] = 0u;

#if USE_TDM
  float* tiles = (float*)(smem + NB_HI);        // 2 x TILE_ELEMS staging (64 KB)
  const bool issuer = (threadIdx.x < 32u);      // wave 0 only (wave32)
  const int ntiles = n / TILE_ELEMS;
  int tile = blockIdx.x;
  int cur = 0;
  __syncthreads();                              // histogram cleared before first DMA use
  if (issuer && tile < ntiles) {
    uint32_t off0 = (uint32_t)(uintptr_t)(const void*)&tiles[0];
    tdm_load_1d(w + (size_t)tile * TILE_ELEMS, off0, TILE_ELEMS);
  }
  for (; tile < ntiles; tile += gridDim.x) {
    if (issuer) __builtin_amdgcn_s_wait_tensorcnt(0);  // tile 'cur' landed in LDS
    __syncthreads();  // publish tile; also retire readers of the other buffer
    int next = tile + (int)gridDim.x;
    if (issuer && next < ntiles) {
      uint32_t offn = (uint32_t)(uintptr_t)(const void*)&tiles[(cur ^ 1) * TILE_ELEMS];
      tdm_load_1d(w + (size_t)next * TILE_ELEMS, offn, TILE_ELEMS);
    }
    const f32x4* lbuf = (const f32x4*)&tiles[cur * TILE_ELEMS];
    for (int j = threadIdx.x; j < TILE_ELEMS / 4; j += blockDim.x) {
      f32x4 v = lbuf[j];                        // ds_load_b128
      atomicAdd(&lh[(__float_as_uint(v.x) & 0x7FFFFFFFu) >> 16], 1u);
      atomicAdd(&lh[(__float_as_uint(v.y) & 0x7FFFFFFFu) >> 16], 1u);
      atomicAdd(&lh[(__float_as_uint(v.z) & 0x7FFFFFFFu) >> 16], 1u);
      atomicAdd(&lh[(__float_as_uint(v.w) & 0x7FFFFFFFu) >> 16], 1u);
    }
    cur ^= 1;
  }
#else
  __syncthreads();
  const f32x4* w4 = (const f32x4*)w;
  int n4 = n >> 2;
  int i = blockIdx.x * blockDim.x + threadIdx.x;
  int s = gridDim.x * blockDim.x;
  for (; i < n4; i += s) {
    __builtin_prefetch(&w4[i + s], 0, 0);
    f32x4 v = __builtin_nontemporal_load(&w4[i]);
    atomicAdd(&lh[(__float_as_uint(v.x) & 0x7FFFFFFFu) >> 16], 1u);
    atomicAdd(&lh[(__float_as_uint(v.y) & 0x7FFFFFFFu) >> 16], 1u);
    atomicAdd(&lh[(__float_as_uint(v.z) & 0x7FFFFFFFu) >> 16], 1u);
    atomicAdd(&lh[(__float_as_uint(v.w) & 0x7FFFFFFFu) >> 16], 1u);
  }
#endif

  __syncthreads();
  for (int b = threadIdx.x; b < NB_HI; b += blockDim.x) {
    uint32_t c = lh[b];
    if (c) atomicAdd(&gh[b], c);  // only ~4k bins are nonzero for normal data
  }
}

// Find descending-rank buckets for both target ranks in the hi histogram.
__global__ __launch_bounds__(256)
void k_select_hi(const uint32_t* __restrict__ hist, uint32_t* __restrict__ ctl,
                 long long rankA, long long rankB) {
  __shared__ uint32_t csum[256];
  const int CH = NB_HI / 256;  // 128 bins per chunk, chunks ordered descending
  int t = threadIdx.x;
  {
    unsigned long long s = 0;
    int hib = NB_HI - 1 - t * CH;
    for (int j = 0; j < CH; ++j) s += hist[hib - j];
    csum[t] = (uint32_t)s;
  }
  __syncthreads();
  if (t == 0) {
    long long ranks[2] = {rankA, rankB};
    for (int r = 0; r < 2; ++r) {
      long long target = ranks[r];
      long long cum = 0;
      uint32_t digit = 0;
      long long rem = 0;
      for (int c = 0; c < 256; ++c) {
        long long cs = (long long)csum[c];
        if (cum + cs > target) {
          int hib = NB_HI - 1 - c * CH;
          for (int j = 0; j < CH; ++j) {
            long long cnt = (long long)hist[hib - j];
            if (cum + cnt > target) {
              digit = (uint32_t)(hib - j);
              rem = target - cum;
              break;
            }
            cum += cnt;
          }
          break;
        }
        cum += cs;
      }
      ctl[2 * r + 0] = digit;
      ctl[2 * r + 1] = (uint32_t)rem;
    }
  }
}

__device__ __forceinline__ void hist_lo_one(float f, uint32_t h0, uint32_t h1,
                                            uint32_t* __restrict__ h2) {
  uint32_t k = __float_as_uint(f) & 0x7FFFFFFFu;
  uint32_t d = k >> 16;
  if (d == h0) atomicAdd(&h2[k & 0xFFFFu], 1u);
  else if (d == h1) atomicAdd(&h2[NB_LO + (k & 0xFFFFu)], 1u);
}

// Pass 2: low-16-bit histogram restricted to the target bucket(s).
// Only ~1e5 of 67M elements match => direct global atomics are fine.
__global__ __launch_bounds__(256)
void k_hist_lo(const f32x4* __restrict__ w4, int n4,
               const uint32_t* __restrict__ ctl, uint32_t* __restrict__ h2) {
  uint32_t h0 = ctl[0];
  uint32_t h1 = ctl[2];
  int i = blockIdx.x * blockDim.x + threadIdx.x;
  int s = gridDim.x * blockDim.x;
  for (; i < n4; i += s) {
    __builtin_prefetch(&w4[i + s], 0, 0);
    f32x4 v = __builtin_nontemporal_load(&w4[i]);
    hist_lo_one(v.x, h0, h1, h2);
    hist_lo_one(v.y, h0, h1, h2);
    hist_lo_one(v.z, h0, h1, h2);
    hist_lo_one(v.w, h0, h1, h2);
  }
}

// Resolve exact 32-bit keys for both order statistics; produce t^2.
__global__ __launch_bounds__(256)
void k_select_lo(const uint32_t* __restrict__ h2, uint32_t* __restrict__ ctl) {
  __shared__ uint32_t csum[2][256];
  const int CH = NB_LO / 256;  // 256
  int t = threadIdx.x;
  uint32_t h0 = ctl[0], h1 = ctl[2];
  const uint32_t* hb = (h0 == h1) ? h2 : (h2 + NB_LO);
  {
    unsigned long long s0 = 0, s1 = 0;
    int hib = NB_LO - 1 - t * CH;
    for (int j = 0; j < CH; ++j) { s0 += h2[hib - j]; s1 += hb[hib - j]; }
    csum[0][t] = (uint32_t)s0;
    csum[1][t] = (uint32_t)s1;
  }
  __syncthreads();
  if (t == 0) {
    uint32_t keys[2];
    for (int r = 0; r < 2; ++r) {
      const uint32_t* H = (r == 0) ? h2 : hb;
      long long target = (long long)ctl[2 * r + 1];
      long long cum = 0;
      uint32_t digit = 0;
      for (int c = 0; c < 256; ++c) {
        long long cs = (long long)csum[r][c];
        if (cum + cs > target) {
          int hib = NB_LO - 1 - c * CH;
          for (int j = 0; j < CH; ++j) {
            long long cnt = (long long)H[hib - j];
            if (cum + cnt > target) {
              digit = (uint32_t)(hib - j);
              break;
            }
            cum += cnt;
          }
          break;
        }
        cum += cs;
      }
      uint32_t hi16 = (r == 0) ? h0 : h1;
      keys[r] = (hi16 << 16) | digit;
    }
    float Wh = __uint_as_float(keys[0]);
    float Wt = __uint_as_float(keys[1]);
    float thr = 0.5f * (Wh + Wt);
    ctl[4] = keys[0];
    ctl[5] = keys[1];
    ctl[6] = __float_as_uint(thr * thr);
  }
}

// Pass 3: out = w * sigmoid((w^2 - t^2)/T) = w / (1 + exp((t^2 - w^2)*100))
__global__ __launch_bounds__(256)
void k_apply(const f32x4* __restrict__ w4, f32x4* __restrict__ o4, int n4,
             const uint32_t* __restrict__ ctl) {
  const float t2 = __uint_as_float(ctl[6]);  // uniform -> scalar load
  int i = blockIdx.x * blockDim.x + threadIdx.x;
  int s = gridDim.x * blockDim.x;
  for (; i < n4; i += s) {
    __builtin_prefetch(&w4[i + s], 0, 0);
    f32x4 v = __builtin_nontemporal_load(&w4[i]);
    f32x4 r;
    r.x = v.x / (1.0f + __expf((t2 - v.x * v.x) * 100.0f));
    r.y = v.y / (1.0f + __expf((t2 - v.y * v.y) * 100.0f));
    r.z = v.z / (1.0f + __expf((t2 - v.z * v.z) * 100.0f));
    r.w = v.w / (1.0f + __expf((t2 - v.w * v.w) * 100.0f));
    __builtin_nontemporal_store(r, &o4[i]);
  }
}

extern "C" void kernel_launch(void* const* d_in, const int* in_sizes, int n_in,
                              void* d_out, int out_size, void* d_ws, size_t ws_size,
                              hipStream_t stream) {
  const float* w = (const float*)d_in[0];
  float* out = (float*)d_out;
  long long n = (long long)in_sizes[0];   // 4096*16384 = 67108864
  int n4 = (int)(n >> 2);

  uint32_t* ws = (uint32_t*)d_ws;
  uint32_t* h1 = ws + WS_H1;
  uint32_t* h2 = ws + WS_H2;
  uint32_t* ctl = ws + WS_CTL;

  // ind = int((1-0.9)*n) - 1, clipped to [0, n-2]; ranks are 0-indexed
  // descending positions of the two adjacent order statistics.
  long long ind = (long long)((1.0 - 0.9) * (double)n) - 1;
  long long lim = ind < 0 ? 0 : ind;
  if (lim > n - 2) lim = n - 2;

  k_zero<<<512, 256, 0, stream>>>(ws, WS_WORDS);
  // Always request the full (TDM-layout) LDS size so host/device agree.
  k_hist_hi<<<512, 512, HIST_SMEM_BYTES, stream>>>(w, (int)n, h1);
  k_select_hi<<<1, 256, 0, stream>>>(h1, ctl, lim, lim + 1);
  k_hist_lo<<<2048, 256, 0, stream>>>((const f32x4*)w, n4, ctl, h2);
  k_select_lo<<<1, 256, 0, stream>>>(h2, ctl);
  k_apply<<<4096, 256, 0, stream>>>((const f32x4*)w, (f32x4*)out, n4, ctl);
}